// EERNNASeqModel_61375082660584
// MI455X (gfx1250) — compile-verified
//
#include <hip/hip_runtime.h>
#include <hip/hip_bf16.h>
#include <math.h>

#define Dd 768
#define Hh 1024
#define Ll 16384

typedef float v2f __attribute__((ext_vector_type(2)));
typedef float v8f __attribute__((ext_vector_type(8)));

// out[r] = dot(M[r, off : off+K], v[0:K]) + bias[r]
// off = (sel==null || sel[0]>=0.5) ? offA : offB   (device-side select for x_r gating)
// One wave -> 16 rows via V_WMMA_F32_16X16X4_F32; K consumed 16/iter with two
// interleaved accumulators to break the WMMA->WMMA RAW chain and keep 4
// global_load_b64 per wave in flight.
// A tile: lane holds row (lane&15), K pair at kb + (lane<16 ? 0 : 2)  [ISA 16x4 f32 A layout]
// B tile: vector broadcast across N: lane holds v[kb+kOff], v[kb+kOff+1]
// D tile: VGPR g -> lane0 = row g, lane16 = row 8+g (at N==0), same value in all N.
template <int K>
__global__ void wmma_gemv_f32(const float* __restrict__ M, int stride,
                              const float* __restrict__ sel, int offA, int offB,
                              const float* __restrict__ v,
                              const float* __restrict__ bias,
                              float* __restrict__ out, int R)
{
    __shared__ float lv[K];
    const int tid = threadIdx.x;
    for (int k = tid; k < K; k += blockDim.x) lv[k] = v[k];
    __syncthreads();

    const int lane = tid & 31;
    const int wave = tid >> 5;
    const int rowBase = (blockIdx.x * (blockDim.x >> 5) + wave) * 16;
    if (rowBase >= R) return;   // wave-uniform: EXEC stays all-ones for WMMA

    const int off  = (sel != nullptr && sel[0] < 0.5f) ? offB : offA;
    const int kOff = (lane < 16) ? 0 : 2;
    const float* mrow = M + (size_t)(rowBase + (lane & 15)) * (size_t)stride + off;

    v8f acc0 = {};
    v8f acc1 = {};
    for (int kb = 0; kb < K; kb += 16) {
        // Batch the 4 A-tile loads (compiler clauses them -> 4 loads in flight)
        v2f a0 = *(const v2f*)(mrow + kb +  0 + kOff);
        v2f a1 = *(const v2f*)(mrow + kb +  4 + kOff);
        v2f a2 = *(const v2f*)(mrow + kb +  8 + kOff);
        v2f a3 = *(const v2f*)(mrow + kb + 12 + kOff);
        v2f b0, b1, b2, b3;
        b0.x = lv[kb +  0 + kOff]; b0.y = lv[kb +  1 + kOff];
        b1.x = lv[kb +  4 + kOff]; b1.y = lv[kb +  5 + kOff];
        b2.x = lv[kb +  8 + kOff]; b2.y = lv[kb +  9 + kOff];
        b3.x = lv[kb + 12 + kOff]; b3.y = lv[kb + 13 + kOff];
        acc0 = __builtin_amdgcn_wmma_f32_16x16x4_f32(false, a0, false, b0, (short)0, acc0, false, false);
        acc1 = __builtin_amdgcn_wmma_f32_16x16x4_f32(false, a1, false, b1, (short)0, acc1, false, false);
        acc0 = __builtin_amdgcn_wmma_f32_16x16x4_f32(false, a2, false, b2, (short)0, acc0, false, false);
        acc1 = __builtin_amdgcn_wmma_f32_16x16x4_f32(false, a3, false, b3, (short)0, acc1, false, false);
    }

    if (lane == 0) {
        #pragma unroll
        for (int g = 0; g < 8; ++g)
            out[rowBase + g] = acc0[g] + acc1[g] + (bias ? bias[rowBase + g] : 0.0f);
    } else if (lane == 16) {
        #pragma unroll
        for (int g = 0; g < 8; ++g)
            out[rowBase + 8 + g] = acc0[g] + acc1[g] + (bias ? bias[rowBase + 8 + g] : 0.0f);
    }
}

// Single block, 1024 threads: softmax over logits in place; also zero-init attn accumulator.
__global__ void softmax_zero_kernel(float* __restrict__ logits, int n,
                                    float* __restrict__ attn, int an)
{
    __shared__ float red[1024];
    const int tid = threadIdx.x;
    if (tid < an) attn[tid] = 0.0f;

    float m = -INFINITY;
    for (int i = tid; i < n; i += 1024) m = fmaxf(m, logits[i]);
    red[tid] = m; __syncthreads();
    for (int s = 512; s > 0; s >>= 1) { if (tid < s) red[tid] = fmaxf(red[tid], red[tid + s]); __syncthreads(); }
    m = red[0]; __syncthreads();

    float sum = 0.0f;
    for (int i = tid; i < n; i += 1024) sum += __expf(logits[i] - m);
    red[tid] = sum; __syncthreads();
    for (int s = 512; s > 0; s >>= 1) { if (tid < s) red[tid] += red[tid + s]; __syncthreads(); }
    const float inv = 1.0f / red[0];

    for (int i = tid; i < n; i += 1024) logits[i] = __expf(logits[i] - m) * inv;
}

// attn[col] += sum_{r in block row-chunk} alpha[r] * hidden[r, col]
// float2 per thread: block covers 512 consecutive cols, 256 rows -> 512KB/block,
// fully coalesced global_load_b64. grid = (L/256, H/512).
__global__ void attn_weighted_sum_kernel(const float* __restrict__ alpha,
                                         const float* __restrict__ hidden,
                                         float* __restrict__ attn)
{
    const int col = (blockIdx.y * 256 + threadIdx.x) * 2;
    const int r0  = blockIdx.x * 256;
    const float* hp = hidden + (size_t)r0 * Hh + col;
    float accx = 0.0f, accy = 0.0f;
    #pragma unroll 4
    for (int r = 0; r < 256; ++r) {
        v2f hv = *(const v2f*)(hp + (size_t)r * Hh);
        const float a = alpha[r0 + r];
        accx = fmaf(a, hv.x, accx);
        accy = fmaf(a, hv.y, accy);
    }
    atomicAdd(&attn[col + 0], accx);
    atomicAdd(&attn[col + 1], accy);
}

// Single block, 1024 threads: score (LDS tree reduce over 1792-elem dot) + GRU combine.
__global__ void gru_score_kernel(const float* __restrict__ gi, const float* __restrict__ gh,
                                 const float* __restrict__ hprev, const float* __restrict__ text_v,
                                 const float* __restrict__ attn, const float* __restrict__ score_W,
                                 const float* __restrict__ score_b, float* __restrict__ out)
{
    __shared__ float red[1024];
    const int tid = threadIdx.x;

    float p = attn[tid] * score_W[Dd + tid];
    if (tid < Dd) p += text_v[tid] * score_W[tid];
    red[tid] = p; __syncthreads();
    for (int s = 512; s > 0; s >>= 1) { if (tid < s) red[tid] += red[tid + s]; __syncthreads(); }
    if (tid == 0) out[0] = red[0] + score_b[0];

    const float r  = 1.0f / (1.0f + __expf(-(gi[tid] + gh[tid])));
    const float z  = 1.0f / (1.0f + __expf(-(gi[Hh + tid] + gh[Hh + tid])));
    const float nn = tanhf(gi[2 * Hh + tid] + r * gh[2 * Hh + tid]);
    out[1 + tid] = (1.0f - z) * nn + z * hprev[tid];
}

extern "C" void kernel_launch(void* const* d_in, const int* in_sizes, int n_in,
                              void* d_out, int out_size, void* d_ws, size_t ws_size,
                              hipStream_t stream) {
    (void)in_sizes; (void)n_in; (void)out_size; (void)ws_size;
    const float* text_v   = (const float*)d_in[0];
    const float* result   = (const float*)d_in[1];
    const float* text_s   = (const float*)d_in[2];
    const float* hidden_s = (const float*)d_in[3];
    const float* W_ih     = (const float*)d_in[4];
    const float* W_hh     = (const float*)d_in[5];
    const float* b_ih     = (const float*)d_in[6];
    const float* b_hh     = (const float*)d_in[7];
    const float* score_W  = (const float*)d_in[8];
    const float* score_b  = (const float*)d_in[9];
    float* out = (float*)d_out;

    float* alpha = (float*)d_ws;        // L       (logits, then softmax weights in place)
    float* attn  = alpha + Ll;          // H
    float* gi    = attn + Hh;           // 3H
    float* gh    = gi + 3 * Hh;         // 3H

    const float* hlast = hidden_s + (size_t)(Ll - 1) * Hh;

    // 1) logits = text_s @ text_v       (16384x768 matvec, WMMA)  256 blocks x 4 waves
    wmma_gemv_f32<Dd><<<Ll / 64, 128, 0, stream>>>(
        text_s, Dd, nullptr, 0, 0, text_v, nullptr, alpha, Ll);

    // 2) gi = x_r @ W_ih.T + b_ih      (3072x768 matvec over selected half of W_ih)
    wmma_gemv_f32<Dd><<<(3 * Hh) / 64, 128, 0, stream>>>(
        W_ih, 2 * Dd, result, /*offA(result>=.5)=*/0, /*offB=*/Dd, text_v, b_ih, gi, 3 * Hh);

    // 3) gh = h @ W_hh.T + b_hh        (3072x1024 matvec)
    wmma_gemv_f32<Hh><<<(3 * Hh) / 64, 128, 0, stream>>>(
        W_hh, Hh, nullptr, 0, 0, hlast, b_hh, gh, 3 * Hh);

    // 4) alpha = softmax(logits); attn = 0
    softmax_zero_kernel<<<1, 1024, 0, stream>>>(alpha, Ll, attn, Hh);

    // 5) attn += alpha.T @ hidden_s    (streams 67MB, float2 loads, atomic f32 accumulation)
    attn_weighted_sum_kernel<<<dim3(Ll / 256, Hh / 512), 256, 0, stream>>>(alpha, hidden_s, attn);

    // 6) score + GRU combine -> out[0..1024]
    gru_score_kernel<<<1, 1024, 0, stream>>>(gi, gh, hlast, text_v, attn, score_W, score_b, out);
}